// MyGraphSAGE_70042326663942
// MI455X (gfx1250) — compile-verified
//
#include <hip/hip_runtime.h>

typedef float v2f __attribute__((ext_vector_type(2)));
typedef float v8f __attribute__((ext_vector_type(8)));

#define CH 64
#define NGRAPH 512

// ---------------- utility kernels ----------------

__global__ void zero_k(float* __restrict__ p, int n) {
  int i = blockIdx.x * blockDim.x + threadIdx.x;
  if (i < n) p[i] = 0.0f;
}

__global__ void count_k(const int* __restrict__ dst, float* __restrict__ cnt, int E) {
  int i = blockIdx.x * blockDim.x + threadIdx.x;
  if (i < E) unsafeAtomicAdd(&cnt[dst[i]], 1.0f);
}

__global__ void inv_k(const float* __restrict__ cnt, float* __restrict__ inv, int n) {
  int i = blockIdx.x * blockDim.x + threadIdx.x;
  if (i < n) inv[i] = 1.0f / fmaxf(cnt[i], 1.0f);
}

// Repack row-major W[out_ch][64] into per-lane B fragments for
// V_WMMA_F32_16X16X4_F32:  B is KxN (4x16); lane<16 holds K=0,1 (N=lane),
// lane>=16 holds K=2,3 (N=lane-16).  packed[((u*16+t)*32+lane)*2 + {0,1}]
__global__ void pack_k(const float* __restrict__ W, float* __restrict__ packed, int ntiles) {
  int i = blockIdx.x * blockDim.x + threadIdx.x;
  int total = ntiles * 16 * 32;
  if (i >= total) return;
  int lane = i & 31;
  int t = (i >> 5) & 15;       // k-tile (K = 4t .. 4t+3)
  int u = i >> 9;              // n-tile
  int n = u * 16 + (lane & 15);
  int k = 4 * t + 2 * (lane >> 4);
  packed[i * 2 + 0] = W[n * 64 + k + 0];
  packed[i * 2 + 1] = W[n * 64 + k + 1];
}

// ---------------- edge scatter (aggregation) ----------------
// one wave per edge; 2 channels per lane; coalesced 256B gather per edge,
// hardware float atomics into the L2-resident accumulator.
__global__ void scatter_k(const float* __restrict__ h, const int* __restrict__ ei,
                          float* __restrict__ agg, int E) {
  int g = blockIdx.x * blockDim.x + threadIdx.x;
  int e = g >> 5;
  if (e >= E) return;
  int lane = g & 31;
  int s = ei[e];        // src
  int d = ei[E + e];    // dst
  const float* xr = h + (size_t)s * CH + 2 * lane;
  float* ar = agg + (size_t)d * CH + 2 * lane;
  unsafeAtomicAdd(&ar[0], xr[0]);
  unsafeAtomicAdd(&ar[1], xr[1]);
}

// ---------------- fused SAGE linear layer ----------------
// out[16 nodes x 64] = (agg * inv) @ Wl^T + x @ Wr^T + b   (+ optional ReLU)
// one wave per 16-node tile, 4 N-tiles per wave, K=64 in steps of 4.
__global__ __launch_bounds__(256) void sage_gemm_k(
    const float* __restrict__ x, const float* __restrict__ agg,
    const float* __restrict__ inv, const float* __restrict__ pWl,
    const float* __restrict__ pWr, const float* __restrict__ bias,
    float* __restrict__ out, int nnodes, int do_relu) {
  int wid = (blockIdx.x * blockDim.x + threadIdx.x) >> 5;
  int base = wid * 16;
  if (base >= nnodes) return;            // whole-wave uniform exit (EXEC all-1s kept)
  int lane = threadIdx.x & 31;
  int m = lane & 15;                     // A row within tile
  int hi = lane >> 4;
  int koff = 2 * hi;                     // A: lanes>=16 carry K=2,3 of each 4-chunk

  float sc = inv[base + m];              // fold mean-division into A load
  v8f acc[4];
#pragma unroll
  for (int u = 0; u < 4; ++u) {          // fold bias into accumulator init
    float b = bias[u * 16 + m];          // D col = lane&15
    acc[u] = (v8f){b, b, b, b, b, b, b, b};
  }

  const float* xrow = x + (size_t)(base + m) * CH + koff;
  const float* arow = agg + (size_t)(base + m) * CH + koff;

#pragma unroll
  for (int t = 0; t < 16; ++t) {
    v2f ax = *(const v2f*)(xrow + 4 * t);
    v2f aa = *(const v2f*)(arow + 4 * t);
    aa.x *= sc;
    aa.y *= sc;
#pragma unroll
    for (int u = 0; u < 4; ++u) {
      v2f bl = *(const v2f*)(pWl + (size_t)((u * 16 + t) * 32 + lane) * 2);
      v2f br = *(const v2f*)(pWr + (size_t)((u * 16 + t) * 32 + lane) * 2);
      acc[u] = __builtin_amdgcn_wmma_f32_16x16x4_f32(false, aa, false, bl,
                                                     (short)0, acc[u], false, false);
      acc[u] = __builtin_amdgcn_wmma_f32_16x16x4_f32(false, ax, false, br,
                                                     (short)0, acc[u], false, false);
    }
  }

  // D layout: VGPR r -> row = r + 8*hi, col = lane&15 (+ 16u)
#pragma unroll
  for (int u = 0; u < 4; ++u) {
#pragma unroll
    for (int r = 0; r < 8; ++r) {
      int row = base + r + 8 * hi;
      float v = acc[u][r];
      if (do_relu) v = fmaxf(v, 0.0f);
      out[(size_t)row * CH + u * 16 + m] = v;
    }
  }
}

// ---------------- graph mean pool scatter ----------------
__global__ void pool_k(const float* __restrict__ h, const int* __restrict__ batch,
                       float* __restrict__ pooled, float* __restrict__ pcnt, int n) {
  int g = blockIdx.x * blockDim.x + threadIdx.x;
  int node = g >> 5;
  if (node >= n) return;
  int lane = g & 31;
  int gr = batch[node];
  const float* hr = h + (size_t)node * CH + 2 * lane;
  unsafeAtomicAdd(&pooled[gr * CH + 2 * lane + 0], hr[0]);
  unsafeAtomicAdd(&pooled[gr * CH + 2 * lane + 1], hr[1]);
  if (lane == 0) unsafeAtomicAdd(&pcnt[gr], 1.0f);
}

// ---------------- final FC (512x64 @ 64x16) via WMMA ----------------
__global__ void fc_k(const float* __restrict__ pooled, const float* __restrict__ pcnt,
                     const float* __restrict__ pWfc, const float* __restrict__ bfc,
                     float* __restrict__ out) {
  int wid = (blockIdx.x * blockDim.x + threadIdx.x) >> 5;
  int base = wid * 16;
  if (base >= NGRAPH) return;
  int lane = threadIdx.x & 31;
  int m = lane & 15;
  int hi = lane >> 4;
  float sc = 1.0f / fmaxf(pcnt[base + m], 1.0f);   // fold pool mean in
  float b = bfc[m];
  v8f acc = (v8f){b, b, b, b, b, b, b, b};
  const float* pr = pooled + (size_t)(base + m) * CH + 2 * hi;
#pragma unroll
  for (int t = 0; t < 16; ++t) {
    v2f a = *(const v2f*)(pr + 4 * t);
    a.x *= sc;
    a.y *= sc;
    v2f bb = *(const v2f*)(pWfc + (size_t)(t * 32 + lane) * 2);
    acc = __builtin_amdgcn_wmma_f32_16x16x4_f32(false, a, false, bb,
                                                (short)0, acc, false, false);
  }
#pragma unroll
  for (int r = 0; r < 8; ++r)
    out[(size_t)(base + r + 8 * hi) * 16 + m] = acc[r];
}

// ---------------- host launcher ----------------

static inline int cdiv(long long a, int b) { return (int)((a + b - 1) / b); }

extern "C" void kernel_launch(void* const* d_in, const int* in_sizes, int n_in,
                              void* d_out, int out_size, void* d_ws, size_t ws_size,
                              hipStream_t stream) {
  (void)n_in; (void)out_size; (void)ws_size;
  const float* x   = (const float*)d_in[0];
  const int*   ei  = (const int*)d_in[1];
  const int*   bat = (const int*)d_in[2];
  const float* W1l = (const float*)d_in[3];
  const float* b1  = (const float*)d_in[4];
  const float* W1r = (const float*)d_in[5];
  const float* W2l = (const float*)d_in[6];
  const float* b2  = (const float*)d_in[7];
  const float* W2r = (const float*)d_in[8];
  const float* W3l = (const float*)d_in[9];
  const float* b3  = (const float*)d_in[10];
  const float* W3r = (const float*)d_in[11];
  const float* Wfc = (const float*)d_in[12];
  const float* bfc = (const float*)d_in[13];
  float* out = (float*)d_out;

  const int N = in_sizes[0] / CH;     // 100000
  const int E = in_sizes[1] / 2;      // 1600000

  // workspace layout (floats)
  float* ws = (float*)d_ws;
  float* cnt = ws;                ws += N;
  float* inv = ws;                ws += N;
  float* agg = ws;                ws += (size_t)N * CH;
  float* hA  = ws;                ws += (size_t)N * CH;
  float* hB  = ws;                ws += (size_t)N * CH;
  float* pk[6];
  for (int i = 0; i < 6; ++i) { pk[i] = ws; ws += 4096; }
  float* pkfc = ws;               ws += 1024;
  float* pooled = ws;             ws += NGRAPH * CH;
  float* pcnt = ws;               ws += NGRAPH;

  const int B = 256;
  const int featN = N * CH;

  // degree counts + reciprocals (edges fixed per call)
  zero_k<<<cdiv(N, B), B, 0, stream>>>(cnt, N);
  count_k<<<cdiv(E, B), B, 0, stream>>>(ei + E, cnt, E);
  inv_k<<<cdiv(N, B), B, 0, stream>>>(cnt, inv, N);

  // pack all weights into WMMA B-fragment layout
  const float* Wmats[6] = {W1l, W1r, W2l, W2r, W3l, W3r};
  for (int i = 0; i < 6; ++i)
    pack_k<<<cdiv(4 * 16 * 32, B), B, 0, stream>>>(Wmats[i], pk[i], 4);
  pack_k<<<cdiv(1 * 16 * 32, B), B, 0, stream>>>(Wfc, pkfc, 1);

  const int scatterBlocks = cdiv((long long)E * 32, B);
  const int gemmBlocks = cdiv((long long)(N / 16) * 32, B);

  // layer 1: x -> hA (ReLU)
  zero_k<<<cdiv(featN, B), B, 0, stream>>>(agg, featN);
  scatter_k<<<scatterBlocks, B, 0, stream>>>(x, ei, agg, E);
  sage_gemm_k<<<gemmBlocks, B, 0, stream>>>(x, agg, inv, pk[0], pk[1], b1, hA, N, 1);

  // layer 2: hA -> hB (ReLU)
  zero_k<<<cdiv(featN, B), B, 0, stream>>>(agg, featN);
  scatter_k<<<scatterBlocks, B, 0, stream>>>(hA, ei, agg, E);
  sage_gemm_k<<<gemmBlocks, B, 0, stream>>>(hA, agg, inv, pk[2], pk[3], b2, hB, N, 1);

  // layer 3: hB -> hA (no ReLU)
  zero_k<<<cdiv(featN, B), B, 0, stream>>>(agg, featN);
  scatter_k<<<scatterBlocks, B, 0, stream>>>(hB, ei, agg, E);
  sage_gemm_k<<<gemmBlocks, B, 0, stream>>>(hB, agg, inv, pk[4], pk[5], b3, hA, N, 0);

  // global mean pool + FC
  zero_k<<<cdiv(NGRAPH * CH, B), B, 0, stream>>>(pooled, NGRAPH * CH);
  zero_k<<<cdiv(NGRAPH, B), B, 0, stream>>>(pcnt, NGRAPH);
  pool_k<<<cdiv((long long)N * 32, B), B, 0, stream>>>(hA, bat, pooled, pcnt, N);
  fc_k<<<cdiv(NGRAPH / 16 * 32, B), B, 0, stream>>>(pooled, pcnt, pkfc, bfc, out);
}